// GraphConvolution_50611894616712
// MI455X (gfx1250) — compile-verified
//
#include <hip/hip_runtime.h>

typedef __attribute__((ext_vector_type(2))) float v2f;
typedef __attribute__((ext_vector_type(8))) float v8f;

#define IN_DIM  96
#define OUT_DIM 96

// ---------------------------------------------------------------------------
// Kernel 1: h = x @ W.T + b   using V_WMMA_F32_16X16X4_F32 (exact fp32 WMMA)
// One wave computes a 16(node) x 16(out) tile; K loop = 96/4 = 24 WMMAs.
// A layout (16x4 f32): lanes 0-15 -> M=lane, v0=K0,v1=K1 ; lanes 16-31 -> v0=K2,v1=K3
// B = W.T (4x16): lane%16 -> N column, lane half selects K pair -> same indexing on W rows.
// C/D (16x16 f32): lane holds col N=lane%16; VGPR r holds row M = r + 8*(lane/16).
// ---------------------------------------------------------------------------
__global__ __launch_bounds__(32) void gconv_gemm_wmma_f32(
    const float* __restrict__ x,   // [N, 96]
    const float* __restrict__ W,   // [96, 96] (out-major)
    const float* __restrict__ bias,// [96]
    float* __restrict__ h,         // [N, 96]
    int n_nodes)
{
  const int lane = threadIdx.x & 31;
  const int half = lane >> 4;        // 0: K={0,1}, 1: K={2,3}
  const int l16  = lane & 15;
  const int tile_m = blockIdx.x;     // node tile (16 nodes)
  const int tile_n = blockIdx.y;     // output-channel tile (16 ch)

  const int row = tile_m * 16 + l16;           // node row fed to A
  const int col = tile_n * 16 + l16;           // out column fed from B (row of W)

  const float* xrow = x + (size_t)row * IN_DIM + 2 * half;
  const float* wrow = W + (size_t)col * IN_DIM + 2 * half;

  v8f acc = {};
#pragma unroll
  for (int k = 0; k < IN_DIM; k += 4) {
    v2f a = *(const v2f*)(xrow + k);   // x[row][k+2h], x[row][k+2h+1]
    v2f b = *(const v2f*)(wrow + k);   // W[col][k+2h], W[col][k+2h+1] == B^T pattern
    acc = __builtin_amdgcn_wmma_f32_16x16x4_f32(
        /*neg_a=*/false, a, /*neg_b=*/false, b,
        /*c_mod=*/(short)0, acc, /*reuse_a=*/false, /*reuse_b=*/false);
  }

  const float bv = bias[col];
  float* hp = h + (size_t)(tile_m * 16 + 8 * half) * OUT_DIM + col;
#pragma unroll
  for (int r = 0; r < 8; ++r) {
    hp[(size_t)r * OUT_DIM] = acc[r] + bv;     // node = tile_m*16 + r + 8*half
  }
}

// ---------------------------------------------------------------------------
// Kernel 2: zero the output accumulator (harness poisons d_out with 0xAA)
// ---------------------------------------------------------------------------
__global__ void gconv_zero_f32(float4* __restrict__ out, int n4)
{
  int i = blockIdx.x * blockDim.x + threadIdx.x;
  if (i < n4) out[i] = make_float4(0.f, 0.f, 0.f, 0.f);
}

// ---------------------------------------------------------------------------
// Kernel 3: COO scatter: out[dst] += w * h[src]
// One wave per edge; lane c handles columns c, c+32, c+64 -> coalesced 384B
// row gather (hits L2: h is 19.2MB of 192MB) + hw fp32 atomics into L2.
// ---------------------------------------------------------------------------
__global__ __launch_bounds__(256) void gconv_scatter(
    const float* __restrict__ h,
    const int*   __restrict__ edge_index,   // [2, E] flat: dst = [0:E), src = [E:2E)
    const float* __restrict__ ew,           // [E]
    float*       __restrict__ out,          // [N, 96]
    int n_edges)
{
  const int lane   = threadIdx.x & 31;
  const int wid    = (blockIdx.x * (blockDim.x >> 5)) + (threadIdx.x >> 5);
  const int nwaves = gridDim.x * (blockDim.x >> 5);

  for (int e = wid; e < n_edges; e += nwaves) {
    const int dst   = edge_index[e];
    const int src   = edge_index[n_edges + e];
    const float w   = ew[e];
    const float* hs = h   + (size_t)src * OUT_DIM;
    float*       od = out + (size_t)dst * OUT_DIM;
#pragma unroll
    for (int c = 0; c < 3; ++c) {
      const int col = lane + 32 * c;
      unsafeAtomicAdd(&od[col], hs[col] * w);   // -> global_atomic_add_f32
    }
  }
}

// ---------------------------------------------------------------------------
extern "C" void kernel_launch(void* const* d_in, const int* in_sizes, int n_in,
                              void* d_out, int out_size, void* d_ws, size_t ws_size,
                              hipStream_t stream)
{
  const float* x    = (const float*)d_in[0];   // [N, 96]
  const int*   ei   = (const int*)  d_in[1];   // [2, E]
  const float* ew   = (const float*)d_in[2];   // [E]
  const float* W    = (const float*)d_in[3];   // [96, 96]
  const float* bias = (const float*)d_in[4];   // [96]
  float* out        = (float*)d_out;           // [N, 96]

  const int n_nodes = in_sizes[0] / IN_DIM;    // 50000
  const int n_edges = in_sizes[2];             // 800000

  float* h = (float*)d_ws;                     // [N, 96] scratch (19.2 MB)

  // Phase 1: dense projection via fp32 WMMA
  {
    dim3 grid((n_nodes + 15) / 16, OUT_DIM / 16);
    gconv_gemm_wmma_f32<<<grid, 32, 0, stream>>>(x, W, bias, h, n_nodes);
  }

  // Phase 2a: zero output accumulator
  {
    const int n4 = (n_nodes * OUT_DIM) / 4;
    gconv_zero_f32<<<(n4 + 255) / 256, 256, 0, stream>>>((float4*)out, n4);
  }

  // Phase 2b: edge scatter with hardware fp32 atomics (L2-resident)
  {
    const int waves_per_block = 256 / 32;
    int blocks = (n_edges + waves_per_block - 1) / waves_per_block;
    gconv_scatter<<<blocks, 256, 0, stream>>>(h, ei, ew, out, n_edges);
  }
}